// SpatialConceptGraph_19198503813137
// MI455X (gfx1250) — compile-verified
//
#include <hip/hip_runtime.h>
#include <math.h>

#define NC 6
#define HH 224
#define WW 224
#define NPIX (HH * WW)        // 50176
#define TAU 0.5f
#define ALPHA 0.2f
#define TWO_SIGMA2 1.28f      // 2 * 0.8^2

typedef float v2f __attribute__((ext_vector_type(2)));
typedef float v8f __attribute__((ext_vector_type(8)));

static __device__ const float c_sign[6] = {1.f, -1.f, 1.f, -1.f, 1.f, 1.f};
// P[c][d] row-major, from _build_prior_matrix()
static __device__ const float c_prior[36] = {
    1.0f, 0.0f, 0.6f, 0.0f, -2.0f, 0.0f,
    0.0f, 1.0f, 0.6f, 0.0f,  0.0f, 0.0f,
    0.1f, 0.1f, 0.5f, 0.0f,  0.0f, 0.0f,
    0.0f, 0.0f, 0.0f, 1.0f,  0.0f, 0.0f,
    0.0f, 0.0f, 0.0f, 0.0f,  1.0f, 0.2f,
    0.0f,-0.6f,-0.6f,-0.6f,  0.6f, 1.0f,
};

// ---------------------------------------------------------------------------
// Kernel 1: per-channel normalize + sigmoid + 32x32 avg pool -> (CH, 49)
// One 256-thread block (8 waves) per (b,c) channel. Memory-bound: 154MB read.
// ---------------------------------------------------------------------------
__global__ __launch_bounds__(256) void k_norm_pool(const float* __restrict__ x,
                                                   float* __restrict__ xs) {
  const int ch = blockIdx.x;
  const float* p = x + (size_t)ch * NPIX;
  const int tid = threadIdx.x;
  const int lane = tid & 31;
  const int wave = tid >> 5;

  // ---- pass 1: sum / sumsq (float4 vectorized) ----
  float s = 0.f, s2 = 0.f;
  const float4* p4 = (const float4*)p;
  for (int i = tid; i < NPIX / 4; i += 256) {
    float4 v = p4[i];
    s  += v.x + v.y + v.z + v.w;
    s2 += v.x * v.x + v.y * v.y + v.z * v.z + v.w * v.w;
  }
  #pragma unroll
  for (int o = 16; o > 0; o >>= 1) {
    s  += __shfl_xor(s, o, 32);
    s2 += __shfl_xor(s2, o, 32);
  }
  __shared__ float rs[8], rs2[8];
  __shared__ float sh_mean, sh_scale;
  __shared__ float pool[49];
  if (lane == 0) { rs[wave] = s; rs2[wave] = s2; }
  if (tid < 49) pool[tid] = 0.f;
  __syncthreads();
  if (tid == 0) {
    float ts = 0.f, ts2 = 0.f;
    #pragma unroll
    for (int i = 0; i < 8; ++i) { ts += rs[i]; ts2 += rs2[i]; }
    float mean = ts / (float)NPIX;
    float var  = (ts2 - (float)NPIX * mean * mean) / (float)(NPIX - 1); // ddof=1
    float stdv = sqrtf(fmaxf(var, 0.f)) + 1e-5f;
    sh_mean  = mean;
    sh_scale = 1.f / (stdv * TAU);
  }
  __syncthreads();
  const float mean = sh_mean, scale = sh_scale;

  // ---- pass 2 (L2-resident re-read): sigmoid + window accumulation ----
  // wave w owns rows == w (mod 8); lane l owns cols l + 32*c7.
  for (int rw = 0; rw < 7; ++rw) {          // row-window
    float acc[7] = {0.f, 0.f, 0.f, 0.f, 0.f, 0.f, 0.f};
    for (int sub = 0; sub < 4; ++sub) {
      const int row = rw * 32 + sub * 8 + wave;
      const float* pr = p + row * WW;
      #pragma unroll
      for (int c7 = 0; c7 < 7; ++c7) {      // col-window
        float v = pr[c7 * 32 + lane];
        float z = (v - mean) * scale;
        acc[c7] += 1.f / (1.f + __expf(-z));
      }
    }
    #pragma unroll
    for (int c7 = 0; c7 < 7; ++c7) {
      float r = acc[c7];
      #pragma unroll
      for (int o = 16; o > 0; o >>= 1) r += __shfl_xor(r, o, 32);
      if (lane == c7) atomicAdd(&pool[rw * 7 + c7], r);   // ds_add_f32
    }
  }
  __syncthreads();
  if (tid < 49) xs[ch * 49 + tid] = pool[tid] * (1.f / 1024.f);
}

// ---------------------------------------------------------------------------
// Kernel 2: salient = relu(sign*(x - mean_s)); S = salient @ Dxx  via
// V_WMMA_F32_16X16X4_F32. One wave per 16-row M tile (48 blocks for 768 rows).
// A: 16x52 (K padded 49->52), B: Dxx 52x64 (N padded 49->64), in LDS.
// ---------------------------------------------------------------------------
__global__ __launch_bounds__(32) void k_salient_gemm(const float* __restrict__ xs,
                                                     float* __restrict__ S,
                                                     int nrows) {
  const int m0 = blockIdx.x * 16;
  const int lane = threadIdx.x;
  __shared__ float A[16][52];
  __shared__ float Bt[52][64];

  // Build Dxx (Gaussian over 7x7 grid), zero-padded to 52x64.
  for (int idx = lane; idx < 52 * 64; idx += 32) {
    int k = idx >> 6, n = idx & 63;
    float v = 0.f;
    if (k < 49 && n < 49) {
      float dy = (float)(k / 7 - n / 7);
      float dx = (float)(k % 7 - n % 7);
      v = __expf(-(dy * dy + dx * dx) / TWO_SIGMA2);
    }
    Bt[k][n] = v;
  }
  // Build A rows: salient = relu(sign * (x_spatial - spatial_mean))
  if (lane < 16) {
    const int row = m0 + lane;
    if (row < nrows) {
      const float* xr = xs + row * 49;
      float sum = 0.f;
      for (int sI = 0; sI < 49; ++sI) sum += xr[sI];
      const float mean = sum * (1.f / 49.f);
      const float sg = c_sign[row % NC];
      for (int sI = 0; sI < 49; ++sI)
        A[lane][sI] = fmaxf(sg * (xr[sI] - mean), 0.f);
    } else {
      for (int sI = 0; sI < 49; ++sI) A[lane][sI] = 0.f;
    }
    A[lane][49] = A[lane][50] = A[lane][51] = 0.f;
  }
  __syncthreads();

  const int l  = lane & 15;
  const int hi = (lane >> 4) & 1;  // lanes 16..31 carry K+2/K+3 halves

  #pragma unroll
  for (int nt = 0; nt < 64; nt += 16) {
    v8f acc = {0.f, 0.f, 0.f, 0.f, 0.f, 0.f, 0.f, 0.f};
    for (int kt = 0; kt < 52; kt += 4) {
      v2f a, b;
      // A 16x4 f32: lanes 0-15 -> {K, K+1}; lanes 16-31 -> {K+2, K+3}
      a[0] = A[l][kt + 2 * hi];
      a[1] = A[l][kt + 2 * hi + 1];
      // B 4x16 f32 (row striped across lanes, mirrored half-wave split)
      b[0] = Bt[kt + 2 * hi][nt + l];
      b[1] = Bt[kt + 2 * hi + 1][nt + l];
      acc = __builtin_amdgcn_wmma_f32_16x16x4_f32(
          /*neg_a=*/false, a, /*neg_b=*/false, b,
          /*c_mod=*/(short)0, acc, /*reuse_a=*/false, /*reuse_b=*/false);
    }
    // D layout: lanes 0-15 -> rows 0..7 col=lane; lanes 16-31 -> rows 8..15
    const int col = nt + l;
    if (col < 49) {
      const int rbase = m0 + hi * 8;
      #pragma unroll
      for (int r = 0; r < 8; ++r) {
        const int row = rbase + r;
        if (row < nrows) S[row * 49 + col] = acc[r];
      }
    }
  }
}

// ---------------------------------------------------------------------------
// Kernel 3: concept mixing (P' = P + 0.2*tanh(Pd)), relu, sign flip, stats.
// One 64-thread block per batch.
// ---------------------------------------------------------------------------
__global__ __launch_bounds__(64) void k_graph_mix(const float* __restrict__ S,
                                                  const float* __restrict__ Pdelta,
                                                  float* __restrict__ gsp,
                                                  float* __restrict__ feat) {
  const int b = blockIdx.x;
  const int tid = threadIdx.x;
  __shared__ float Pm[36];
  __shared__ float G[6][49];
  if (tid < 36) Pm[tid] = c_prior[tid] + ALPHA * tanhf(Pdelta[tid]);
  __syncthreads();
  if (tid < 49) {
    float s6[6];
    #pragma unroll
    for (int c = 0; c < 6; ++c) s6[c] = S[(b * 6 + c) * 49 + tid];
    #pragma unroll
    for (int d = 0; d < 6; ++d) {
      float m = 0.f;
      #pragma unroll
      for (int c = 0; c < 6; ++c) m += Pm[c * 6 + d] * s6[c];
      m = fmaxf(m, 0.f);                 // relu(message)
      const float g = m * c_sign[d];     // sign-flip inverse concepts
      G[d][tid] = g;
      gsp[(b * 6 + d) * 49 + tid] = g;
    }
  }
  __syncthreads();
  if (tid < 18) {
    const int d = tid % 6, which = tid / 6;
    float sm = 0.f, mx = G[d][0], mn = G[d][0];
    for (int t = 0; t < 49; ++t) {
      const float v = G[d][t];
      sm += v; mx = fmaxf(mx, v); mn = fminf(mn, v);
    }
    const float outv = (which == 0) ? sm * (1.f / 49.f) : (which == 1) ? mx : mn;
    feat[b * 18 + which * 6 + d] = outv;
  }
}

// ---------------------------------------------------------------------------
// Kernel 4: bilinear 7x7 -> 224x224 upsample (half-pixel centers, edge clamp).
// One 256-thread block per channel; fully coalesced float4 stores (154MB).
// ---------------------------------------------------------------------------
__global__ __launch_bounds__(256) void k_upsample(const float* __restrict__ gsp,
                                                  float* __restrict__ out) {
  const int ch = blockIdx.x;
  const int tid = threadIdx.x;
  __shared__ float g[49];
  __shared__ int   i0s[224], i1s[224];
  __shared__ float frs[224];
  if (tid < 49) g[tid] = gsp[ch * 49 + tid];
  if (tid < 224) {
    const float src = ((float)tid + 0.5f) * (1.f / 32.f) - 0.5f;
    const float fl = floorf(src);
    const int a0 = (int)fl;
    frs[tid] = src - fl;
    i0s[tid] = a0 < 0 ? 0 : (a0 > 6 ? 6 : a0);
    const int a1 = a0 + 1;
    i1s[tid] = a1 < 0 ? 0 : (a1 > 6 ? 6 : a1);
  }
  __syncthreads();
  float4* op = (float4*)(out + (size_t)ch * NPIX);
  for (int q = tid; q < NPIX / 4; q += 256) {
    const int h  = q / 56;
    const int w0 = (q % 56) * 4;
    const int h0 = i0s[h], h1 = i1s[h];
    const float fh = frs[h];
    float4 r;
    float* rp = &r.x;
    #pragma unroll
    for (int j = 0; j < 4; ++j) {
      const int w = w0 + j;
      const int c0 = i0s[w], c1 = i1s[w];
      const float fw = frs[w];
      const float top = g[h0 * 7 + c0] * (1.f - fw) + g[h0 * 7 + c1] * fw;
      const float bot = g[h1 * 7 + c0] * (1.f - fw) + g[h1 * 7 + c1] * fw;
      rp[j] = top * (1.f - fh) + bot * fh;
    }
    op[q] = r;
  }
}

// ---------------------------------------------------------------------------
extern "C" void kernel_launch(void* const* d_in, const int* in_sizes, int n_in,
                              void* d_out, int out_size, void* d_ws, size_t ws_size,
                              hipStream_t stream) {
  const float* x      = (const float*)d_in[0];   // (B, 6, 224, 224) fp32
  const float* Pdelta = (const float*)d_in[1];   // (6, 6) fp32
  float* out = (float*)d_out;

  const int B  = in_sizes[0] / (NC * NPIX);      // 128
  const int CH = B * NC;                         // 768

  float* ws   = (float*)d_ws;
  float* xs   = ws;                 // CH*49 pooled probabilities
  float* S    = ws + (size_t)CH * 49;       // CH*49 salient @ Dxx
  float* gsp  = ws + (size_t)2 * CH * 49;   // CH*49 mixed graph
  float* feat = out;                        // (B, 18)
  float* img  = out + (size_t)B * 18;       // (B, 6, 224, 224)

  k_norm_pool   <<<CH, 256, 0, stream>>>(x, xs);
  k_salient_gemm<<<(CH + 15) / 16, 32, 0, stream>>>(xs, S, CH);
  k_graph_mix   <<<B, 64, 0, stream>>>(S, Pdelta, gsp, feat);
  k_upsample    <<<CH, 256, 0, stream>>>(gsp, img);
}